// VectorQuantizer_18511309046214
// MI455X (gfx1250) — compile-verified
//
#include <hip/hip_runtime.h>

// CDNA5 (gfx1250) VQ-VAE vector quantizer using V_WMMA_F32_16X16X4_F32.
// Distances: argmin_k (||e_k||^2 - 2 z.e_k)  (||z||^2 dropped: row-constant).

typedef __attribute__((ext_vector_type(2))) float v2f;
typedef __attribute__((ext_vector_type(8))) float v8f;

#define NUM_K 512
#define CDIM  64
#define HWSZ  4096                   // H*W
#define N_TOT (32 * HWSZ)            // 131072 rows
#define NELEM (N_TOT * CDIM)         // 8388608 elements of q_out
#define LOSS_SCALE (1.25f / 8388608.f)  // (1 + beta) / (N*C)

// Dynamic LDS layout: codebook [512*64] | e2 [512] | per-wave idx [8*16]
#define LDS_FLOATS (NUM_K * CDIM + NUM_K + 8 * 16)

__global__ __launch_bounds__(256) void vq_wmma_kernel(
    const float* __restrict__ z,      // [32, 64, 64, 64] = [B, C, H, W]
    const float* __restrict__ cb,     // [512, 64]
    float* __restrict__ d_loss,       // [1]
    float* __restrict__ d_q,          // [32, 64, 64, 64]
    float* __restrict__ d_idx)        // [131072] (indices stored as float)
{
  extern __shared__ float smem[];
  float* cb_s  = smem;                       // 512 x 64, row major
  float* e2_s  = smem + NUM_K * CDIM;        // ||e_k||^2
  int*   idx_s = (int*)(smem + NUM_K * CDIM + NUM_K);  // 8 waves * 16 rows

  const int tid  = threadIdx.x;
  const int wave = tid >> 5;
  const int lane = tid & 31;
  const int l16  = lane & 15;       // row within half-wave / N column
  const int hi   = lane >> 4;       // half-wave select

  // ---- Stage codebook into LDS (coalesced), then compute ||e_k||^2 in-LDS.
  for (int i = tid; i < NUM_K * CDIM; i += 256) cb_s[i] = cb[i];
  __syncthreads();
  for (int k = tid; k < NUM_K; k += 256) {
    float s = 0.f;
    for (int c = 0; c < CDIM; ++c) {
      const int cc = (c + tid) & (CDIM - 1);   // rotate start: bank-conflict free
      const float v = cb_s[k * CDIM + cc];
      s += v * v;
    }
    e2_s[k] = s;
  }
  __syncthreads();

  // 256 blocks * 8 waves * 4 iters = 8192 tiles of 16 rows = 131072 rows.
  for (int iter = 0; iter < 4; ++iter) {
    const int tile = (blockIdx.x * 8 + wave) * 4 + iter;
    const int n0   = tile * 16;
    const int b    = n0 >> 12;            // batch (tiles never straddle b)
    const int hw0  = n0 & (HWSZ - 1);

    // ---- Load A tile: z rows n0..n0+15, all 64 channels, as 16 K-steps.
    // ISA 32-bit A 16x4 layout: lane l, VGPR v holds A[M=l%16, K=2*(l/16)+v].
    // z element (n, c) lives at z[(b*64 + c)*4096 + hw]; lanes 0-15 are
    // consecutive hw -> 64B-coalesced per half-wave.
    v2f a[16];
    const float* zb = z + ((size_t)b * CDIM) * HWSZ + hw0 + l16;
    #pragma unroll
    for (int s = 0; s < 16; ++s) {
      const int c0 = 4 * s + 2 * hi;
      a[s].x = zb[(size_t)c0 * HWSZ];
      a[s].y = zb[(size_t)(c0 + 1) * HWSZ];
    }

    float best[8];
    int   bidx[8];
    #pragma unroll
    for (int j = 0; j < 8; ++j) { best[j] = 3.4e38f; bidx[j] = 0; }

    // ---- Sweep all 512 codes in 16-wide k-tiles.
    for (int kt = 0; kt < 32; ++kt) {
      const int krow = kt * 16 + l16;     // this lane's code column
      const float* cbr = cb_s + krow * CDIM + 2 * hi;
      v8f acc = {};
      #pragma unroll
      for (int s = 0; s < 16; ++s) {
        // B 4x16 fragment (mirror of A layout): lane l, VGPR v holds
        // B[K=2*(l/16)+v, N=l%16] = codebook[krow][4s + 2*(l/16) + v].
        v2f bv;
        bv.x = cbr[4 * s];
        bv.y = cbr[4 * s + 1];
        acc = __builtin_amdgcn_wmma_f32_16x16x4_f32(
            /*neg_a=*/false, a[s], /*neg_b=*/false, bv,
            /*c_mod=*/(short)0, acc, /*reuse_a=*/false, /*reuse_b=*/false);
      }
      // C/D layout: lane l, VGPR j holds D[M = j + 8*(l/16), N = l%16].
      const float e2v = e2_s[krow];
      #pragma unroll
      for (int j = 0; j < 8; ++j) {
        const float d = e2v - 2.0f * acc[j];
        if (d < best[j]) { best[j] = d; bidx[j] = krow; }
      }
    }

    // ---- Argmin across the 16 lanes of each half-wave (tie -> smaller k).
    #pragma unroll
    for (int off = 1; off < 16; off <<= 1) {
      #pragma unroll
      for (int j = 0; j < 8; ++j) {
        const float ob = __shfl_xor(best[j], off, 32);
        const int   oi = __shfl_xor(bidx[j], off, 32);
        if (ob < best[j] || (ob == best[j] && oi < bidx[j])) {
          best[j] = ob; bidx[j] = oi;
        }
      }
    }
    // Lane 0 holds winners for M=0..7, lane 16 for M=8..15.
    if (l16 == 0) {
      #pragma unroll
      for (int j = 0; j < 8; ++j) idx_s[wave * 16 + hi * 8 + j] = bidx[j];
    }
    // Same-wave DS ops are in-order; make the cross-lane LDS handoff explicit.
    asm volatile("s_wait_dscnt 0" ::: "memory");

    if (lane < 16)
      d_idx[n0 + lane] = (float)idx_s[wave * 16 + lane];

    // ---- Emit quantized output + loss, reusing the A registers (no re-read
    // of z). Same (c, n) mapping as the A load -> coalesced NT stores.
    const int myidx = idx_s[wave * 16 + l16];
    const size_t basep = ((size_t)b * CDIM) * HWSZ + hw0 + l16;
    float lsum = 0.f;
    #pragma unroll
    for (int s = 0; s < 16; ++s) {
      #pragma unroll
      for (int v = 0; v < 2; ++v) {
        const int c = 4 * s + 2 * hi + v;
        const float q = cb_s[myidx * CDIM + c];
        __builtin_nontemporal_store(q, d_q + basep + (size_t)c * HWSZ);
        const float diff = q - (v == 0 ? a[s].x : a[s].y);
        lsum += diff * diff;
      }
    }
    #pragma unroll
    for (int off = 16; off >= 1; off >>= 1) lsum += __shfl_xor(lsum, off, 32);
    if (lane == 0) atomicAdd(d_loss, lsum * LOSS_SCALE);
  }
}

extern "C" void kernel_launch(void* const* d_in, const int* in_sizes, int n_in,
                              void* d_out, int out_size, void* d_ws, size_t ws_size,
                              hipStream_t stream) {
  const float* z  = (const float*)d_in[0];
  const float* cb = (const float*)d_in[1];
  float* out = (float*)d_out;
  // d_out layout: [0] loss | [1 .. 1+8388608) q_out | tail: indices-as-float.
  float* d_loss = out;
  float* d_q    = out + 1;
  float* d_idx  = out + 1 + (size_t)NELEM;

  const size_t shmem = (size_t)LDS_FLOATS * sizeof(float);  // 133,632 B
  (void)hipFuncSetAttribute((const void*)vq_wmma_kernel,
                            hipFuncAttributeMaxDynamicSharedMemorySize,
                            (int)shmem);
  (void)hipMemsetAsync(d_loss, 0, sizeof(float), stream);   // capture-safe
  vq_wmma_kernel<<<256, 256, shmem, stream>>>(z, cb, d_loss, d_q, d_idx);
}